// MeanNegDot_Loss_69320772158191
// MI455X (gfx1250) — compile-verified
//
#include <hip/hip_runtime.h>
#include <math.h>

typedef float v2f __attribute__((ext_vector_type(2)));
typedef float v4f __attribute__((ext_vector_type(4)));
typedef float v8f __attribute__((ext_vector_type(8)));

#define B_TOTAL     8192
#define D           256
#define BLOCK_ROWS  128           // rows per workgroup (8 waves x 16 rows)
#define COL_TILE    128           // columns per LDS-staged iteration
#define NSUB        (COL_TILE/16) // 16x16 subtiles per wave per iteration
#define LDA         260           // padded LDS row stride in floats (260%64=4 -> conflict-free)
#define INV_NEG_T   (-1.0f/0.07f)

// dynamic LDS layout: sA[128*260] | sB[128*260] | sLab[128] | sLoss[16]
#define SMEM_FLOATS (BLOCK_ROWS*LDA + COL_TILE*LDA)
#define SMEM_BYTES  (SMEM_FLOATS*4 + COL_TILE*4 + 16*4)

__global__ __launch_bounds__(256)
void supcon_rowstats_kernel(const float* __restrict__ feat,
                            const int*  __restrict__ labels,
                            float* __restrict__ partial)
{
    extern __shared__ float smem[];
    float* sA    = smem;                       // BLOCK_ROWS x LDA
    float* sB    = smem + BLOCK_ROWS * LDA;    // COL_TILE x LDA
    int*   sLab  = (int*)(smem + SMEM_FLOATS); // COL_TILE labels
    float* sLoss = (float*)(sLab + COL_TILE);  // 16 per-(wave,half) slots

    const int tid  = threadIdx.x;
    const int lane = tid & 31;
    const int wave = tid >> 5;      // 0..7
    const int n    = lane & 15;     // column-within-tile (B/C layout)
    const int hi   = lane >> 4;     // 0/1: K-slot half (A/B), row-half (C)

    const int rowBase = blockIdx.x * BLOCK_ROWS;

    // ---- stage the A row-block into LDS (128 rows x 256 f32) ----
    for (int idx = tid; idx < BLOCK_ROWS * (D / 4); idx += 256) {
        const int r  = idx >> 6;   // D/4 == 64
        const int c4 = idx & 63;
        const float4 v = ((const float4*)(feat + (size_t)(rowBase + r) * D))[c4];
        *(float4*)(sA + r * LDA + c4 * 4) = v;
    }

    // per-lane row labels: C-layout row m = r + 8*hi  (r = 0..7)
    const int waveRowBase = rowBase + wave * 16 + hi * 8;
    int rowLab[8];
    #pragma unroll
    for (int r = 0; r < 8; ++r) rowLab[r] = labels[waveRowBase + r];

    // online per-(lane,row) stats over this lane's column subset
    float lm[8], sneg[8], spos[8], psum[8], npos[8];
    #pragma unroll
    for (int r = 0; r < 8; ++r) {
        lm[r] = -3.4e38f; sneg[r] = 0.f; spos[r] = 0.f; psum[r] = 0.f; npos[r] = 0.f;
    }

    // Fragment bases: each lane loads an 8-wide K chunk as one b128:
    //   half hi takes floats [k + 4*hi .. k + 4*hi + 3].
    // WMMA #1 uses .xy of both halves -> K = {k,k+1,k+4,k+5}
    // WMMA #2 uses .zw of both halves -> K = {k+2,k+3,k+6,k+7}
    const float* aBase  = sA + (wave * 16 + n) * LDA + 4 * hi;
    const float* bBase0 = sB + n * LDA + 4 * hi;            // subtiles 0..3
    const float* bBase4 = bBase0 + 4 * 16 * LDA;            // subtiles 4..7 (keeps DS offsets <64KB)
    __syncthreads();

    for (int cb = 0; cb < B_TOTAL; cb += COL_TILE) {
        // ---- stage column block (128 rows x 256 f32) + its labels ----
        for (int idx = tid; idx < COL_TILE * (D / 4); idx += 256) {
            const int r  = idx >> 6;
            const int c4 = idx & 63;
            const float4 v = ((const float4*)(feat + (size_t)(cb + r) * D))[c4];
            *(float4*)(sB + r * LDA + c4 * 4) = v;
        }
        if (tid < COL_TILE) sLab[tid] = labels[cb + tid];

        // prefetch next column block into WGP$ while this one computes
        const int ncb = cb + COL_TILE;
        if (ncb < B_TOTAL) {
            const char* p = (const char*)(feat + (size_t)ncb * D) + tid * 512;
            __builtin_prefetch(p,       0, 1);
            __builtin_prefetch(p + 128, 0, 1);
            __builtin_prefetch(p + 256, 0, 1);
            __builtin_prefetch(p + 384, 0, 1);
        }
        __syncthreads();

        // ---- 16x128 tile GEMM: 8 accumulators, K = 256 in chunks of 8 ----
        v8f acc[NSUB];
        #pragma unroll
        for (int st = 0; st < NSUB; ++st) acc[st] = (v8f){0,0,0,0,0,0,0,0};

        for (int k = 0; k < D; k += 8) {
            const v4f a = *(const v4f*)(aBase + k);
            const v2f aLo = __builtin_shufflevector(a, a, 0, 1);
            const v2f aHi = __builtin_shufflevector(a, a, 2, 3);
            v4f b[NSUB];
            #pragma unroll
            for (int st = 0; st < NSUB; ++st) {
                const float* bp = (st < 4) ? (bBase0 + st * 16 * LDA)
                                           : (bBase4 + (st - 4) * 16 * LDA);
                b[st] = *(const v4f*)(bp + k);
                const v2f bLo = __builtin_shufflevector(b[st], b[st], 0, 1);
                acc[st] = __builtin_amdgcn_wmma_f32_16x16x4_f32(
                    false, aLo, false, bLo, (short)0, acc[st], false, false);
            }
            #pragma unroll
            for (int st = 0; st < NSUB; ++st) {
                const v2f bHi = __builtin_shufflevector(b[st], b[st], 2, 3);
                acc[st] = __builtin_amdgcn_wmma_f32_16x16x4_f32(
                    false, aHi, false, bHi, (short)0, acc[st], false, false);
            }
        }

        // ---- fused epilogue: online softmax-style row statistics ----
        #pragma unroll
        for (int st = 0; st < NSUB; ++st) {
            const int col  = cb + st * 16 + n;
            const int clab = sLab[st * 16 + n];
            #pragma unroll
            for (int r = 0; r < 8; ++r) {
                const int   row = waveRowBase + r;
                const float aV  = acc[st][r] * INV_NEG_T;       // logits pre-max
                const float nm  = fmaxf(lm[r], aV);             // ref max includes diagonal
                const float sc  = __expf(lm[r] - nm);
                const float e   = __expf(aV - nm);
                sneg[r] *= sc; spos[r] *= sc;
                const bool self = (col == row);
                const bool pos  = (!self) && (clab == rowLab[r]);
                if (pos)        { spos[r] += e; psum[r] += aV; npos[r] += 1.0f; }
                else if (!self) { sneg[r] += e; }
                lm[r] = nm;
            }
        }
        __syncthreads();
    }

    // ---- merge the 16 lanes that share each row (max-rescaled) ----
    #pragma unroll
    for (int r = 0; r < 8; ++r) {
        float m = lm[r], sn = sneg[r], sp = spos[r], ps = psum[r], np = npos[r];
        #pragma unroll
        for (int off = 1; off < 16; off <<= 1) {
            const float om  = __shfl_xor(m,  off, 32);
            const float osn = __shfl_xor(sn, off, 32);
            const float osp = __shfl_xor(sp, off, 32);
            const float ops = __shfl_xor(ps, off, 32);
            const float onp = __shfl_xor(np, off, 32);
            const float nm2 = fmaxf(m, om);
            const float s1 = __expf(m - nm2), s2 = __expf(om - nm2);
            sn = sn * s1 + osn * s2;
            sp = sp * s1 + osp * s2;
            m  = nm2; ps += ops; np += onp;
        }
        lm[r] = m; sneg[r] = sn; spos[r] = sp; psum[r] = ps; npos[r] = np;
    }

    // ---- per-row loss on lane n==0, deterministic per-(wave,half) slots ----
    if (n == 0) {
        float lsum = 0.0f;
        #pragma unroll
        for (int r = 0; r < 8; ++r) {
            const float np    = npos[r];
            const float nneg  = (float)(B_TOTAL - 1) - np;
            const float mNeg  = sneg[r] / (nneg + 1e-8f);
            const float den   = spos[r] + mNeg;
            const float npAdj = (np < 1e-6f) ? 1.0f : np;
            const float mlpp  = (psum[r] - np * lm[r] - np * logf(den)) / npAdj;
            lsum += mlpp;  // TEMPERATURE / BASE_TEMPERATURE == 1.0
        }
        sLoss[wave * 2 + hi] = lsum;
    }
    __syncthreads();
    if (tid == 0) {
        float t = 0.0f;
        #pragma unroll
        for (int i = 0; i < 16; ++i) t += sLoss[i];
        partial[blockIdx.x] = t;
    }
}

__global__ __launch_bounds__(64)
void supcon_reduce_kernel(const float* __restrict__ partial, float* __restrict__ out, int nPart)
{
    __shared__ float s[64];
    const int tid = threadIdx.x;
    s[tid] = (tid < nPart) ? partial[tid] : 0.0f;
    __syncthreads();
    for (int off = 32; off > 0; off >>= 1) {
        if (tid < off) s[tid] += s[tid + off];
        __syncthreads();
    }
    if (tid == 0) out[0] = s[0] / (float)B_TOTAL;
}

extern "C" void kernel_launch(void* const* d_in, const int* in_sizes, int n_in,
                              void* d_out, int out_size, void* d_ws, size_t ws_size,
                              hipStream_t stream)
{
    (void)in_sizes; (void)n_in; (void)out_size; (void)ws_size;
    const float* feat   = (const float*)d_in[0];
    const int*   labels = (const int*)d_in[1];
    // d_in[2] = sigma (unused by the reference computation)
    float* partial = (float*)d_ws;
    float* out     = (float*)d_out;

    const int nBlocks = B_TOTAL / BLOCK_ROWS; // 64
    supcon_rowstats_kernel<<<nBlocks, 256, SMEM_BYTES, stream>>>(feat, labels, partial);
    supcon_reduce_kernel<<<1, 64, 0, stream>>>(partial, out, nBlocks);
}